// TrajectoryDecoder_87170656239737
// MI455X (gfx1250) — compile-verified
//
#include <hip/hip_runtime.h>
#include <math.h>

typedef __attribute__((ext_vector_type(2))) float v2f;
typedef __attribute__((ext_vector_type(8))) float v8f;

#define NBIN     48
#define RADIUS_F 40.0f
#define TWO_PI_F 6.28318530717958647692f

// ---------------------------------------------------------------------------
// Polar bin helper: given rel = src - dst and distance d, produce the 4
// bilinear corner bins and their weights (without valid/win/cnt factors).
// ---------------------------------------------------------------------------
struct Corners {
    int   bin[4];
    float cw[4];
    float win;
};

__device__ inline Corners polar_corners(float rx, float ry, float d) {
    Corners c;
    float q  = d * (1.0f / RADIUS_F);
    float w0 = fmaxf(1.0f - q * q, 0.0f);
    c.win = w0 * w0 * w0;
    float u  = fminf(q, 1.0f) * 2.0f;           // (R-1) = 2
    int   r0 = (int)floorf(u); r0 = r0 > 2 ? 2 : r0;
    float wr = u - (float)r0;
    int   r1 = r0 + 1 > 2 ? 2 : r0 + 1;
    float th = atan2f(ry, rx) * (1.0f / TWO_PI_F);
    th = th - floorf(th);                        // [0,1)
    float tv = th * 16.0f;
    float tf = floorf(tv);
    float wt = tv - tf;
    int   t0 = ((int)tf) & 15;
    int   t1 = (t0 + 1) & 15;
    c.bin[0] = r0 * 16 + t0; c.cw[0] = (1.0f - wr) * (1.0f - wt);
    c.bin[1] = r0 * 16 + t1; c.cw[1] = (1.0f - wr) * wt;
    c.bin[2] = r1 * 16 + t0; c.cw[2] = wr * (1.0f - wt);
    c.bin[3] = r1 * 16 + t1; c.cw[3] = wr * wt;
    return c;
}

// ---------------------------------------------------------------------------
// init: copy p and feat into mutable workspace buffers
// ---------------------------------------------------------------------------
__global__ void k_init(const float* __restrict__ p, const float* __restrict__ feat,
                       float* __restrict__ pbuf, float* __restrict__ featb) {
    int t = blockIdx.x * blockDim.x + threadIdx.x;
    if (t < 1024)  pbuf[t]  = p[t];
    if (t < 98304) featb[t] = feat[t];
}

// ---------------------------------------------------------------------------
// Build B-operand in WMMA fragment layout.
// Logical B matrix: K2[(bin,i,j),(o,m)] = bin<48 ? K[bin,o,i,(m-j)%8]
//                                                : Wd[o,i,(m-j)%8]
// Fragment layout: Bf[((ntile*ksteps + kk)*32 + lane)*2 + slot]
//   k = 4*kk + (lane<16 ? 0 : 2) + slot ; col = ntile*16 + (lane&15)
// ---------------------------------------------------------------------------
__global__ void k_bfrag(const float* __restrict__ K, const float* __restrict__ Wd,
                        float* __restrict__ Bf, int I, int O, int ksteps, int total) {
    int t = blockIdx.x * blockDim.x + threadIdx.x;
    if (t >= total) return;
    int lane  = t & 31;
    int kk    = (t >> 5) % ksteps;
    int ntile = t / (32 * ksteps);
    int kbase = kk * 4 + ((lane >= 16) ? 2 : 0);
    int col   = ntile * 16 + (lane & 15);
    size_t obase = ((size_t)(ntile * ksteps + kk) * 32 + lane) * 2;
    for (int slot = 0; slot < 2; ++slot) {
        int   k = kbase + slot;
        float v = 0.0f;
        if (col < O * 8) {
            int bin = k / (I * 8);
            int rem = k - bin * (I * 8);
            int i = rem >> 3, j = rem & 7;
            int o = col >> 3, m = col & 7;
            int l = (m - j) & 7;
            v = (bin < NBIN) ? K[(((size_t)(bin * O + o)) * I + i) * 8 + l]
                             : Wd[((size_t)(o * I + i)) * 8 + l];
        }
        Bf[obase + slot] = v;
    }
}

// ---------------------------------------------------------------------------
// A-fragment scatter helper: write one destination row's 49*I*8 values into
// the WMMA fragment layout used by k_gemm.
// ---------------------------------------------------------------------------
template <int I>
__device__ inline void store_afrag(float* __restrict__ Af, int bn, int tid,
                                   const float* __restrict__ acc, float selfv) {
    const int BD  = I * 8;
    const int KST = 49 * I * 2;
    int mtile = bn >> 4, lr = bn & 15;
    size_t mbase = (size_t)mtile * KST * 64;
    for (int bin = 0; bin <= NBIN; ++bin) {
        float v = (bin < NBIN) ? acc[bin * BD + tid] : selfv;
        int cidx = bin * BD + tid;
        int kk = cidx >> 2, rem = cidx & 3;
        int lane = lr + ((rem >= 2) ? 16 : 0);
        int slot = rem & 1;
        Af[mbase + ((size_t)kk * 32 + lane) * 2 + slot] = v;
    }
}

// ---------------------------------------------------------------------------
// Layer-0 aggregation with fused pair-weight computation.
// One WG (192 threads) per destination (b,n). Threads 0..63 compute the
// bilinear polar-bin weights for all 64 sources (shared by layers 1/2 via
// global pw/pbin); then every thread owns feature element (i,j) and
// accumulates weighted source features per bin in an exclusive LDS column.
// Bin 48 = destination's own features (equilinear term).
// ---------------------------------------------------------------------------
__global__ void k_agg0(const float* __restrict__ src, const float* __restrict__ p,
                       const float* __restrict__ mask,
                       float4* __restrict__ pw, unsigned int* __restrict__ pbin,
                       float* __restrict__ Af) {
    const int BD = 24 * 8;
    __shared__ float  acc[NBIN * BD];
    __shared__ float4 pwl[64];
    __shared__ unsigned int pbl[64];
    __shared__ float  red[64];
    int bn = blockIdx.x;
    int b = bn >> 6, n = bn & 63;
    int tid = threadIdx.x;

    float rx = 0.f, ry = 0.f, d = 0.f, valid = 0.f;
    if (tid < 64) {
        int s = tid;
        float pdx = p[bn * 2], pdy = p[bn * 2 + 1];
        rx = p[(b * 64 + s) * 2]     - pdx;
        ry = p[(b * 64 + s) * 2 + 1] - pdy;
        d = sqrtf(rx * rx + ry * ry + 1e-12f);
        valid = (d < RADIUS_F ? 1.0f : 0.0f) * mask[b * 64 + s];
        red[s] = valid;
    }
    for (int k = tid; k < NBIN * BD; k += BD) acc[k] = 0.0f;
    __syncthreads();
    if (tid < 64) {
        int s = tid;
        float total = 0.0f;
        for (int k = 0; k < 64; ++k) total += red[k];
        float cnt = fmaxf(total, 1.0f);
        Corners c = polar_corners(rx, ry, d);
        float base = valid * c.win / cnt;
        float4 w;
        w.x = base * c.cw[0]; w.y = base * c.cw[1];
        w.z = base * c.cw[2]; w.w = base * c.cw[3];
        unsigned int pb = (unsigned)c.bin[0] | ((unsigned)c.bin[1] << 8) |
                          ((unsigned)c.bin[2] << 16) | ((unsigned)c.bin[3] << 24);
        pwl[s] = w; pbl[s] = pb;
        pw[bn * 64 + s] = w; pbin[bn * 64 + s] = pb;
    }
    __syncthreads();

    float selfv = 0.0f;
    for (int s = 0; s < 64; ++s) {
        float fv = src[(size_t)(b * 64 + s) * BD + tid];
        if (s == n) selfv = fv;
        float4 w = pwl[s];
        unsigned int pb = pbl[s];
        acc[(pb & 63u) * BD + tid]         += w.x * fv;
        acc[((pb >> 8) & 63u) * BD + tid]  += w.y * fv;
        acc[((pb >> 16) & 63u) * BD + tid] += w.z * fv;
        acc[((pb >> 24) & 63u) * BD + tid] += w.w * fv;
    }
    store_afrag<24>(Af, bn, tid, acc, selfv);
}

// ---------------------------------------------------------------------------
// Generic aggregation for layers 1/2 (I = 8, relu on load), reads pair
// weights produced by k_agg0.
// ---------------------------------------------------------------------------
template <int I>
__global__ void k_agg(const float* __restrict__ src,
                      const float4* __restrict__ pw,
                      const unsigned int* __restrict__ pbin,
                      float* __restrict__ Af) {
    const int BD = I * 8;
    __shared__ float acc[NBIN * BD];
    int bn = blockIdx.x;
    int b = bn >> 6, n = bn & 63;
    int tid = threadIdx.x;
    for (int k = tid; k < NBIN * BD; k += BD) acc[k] = 0.0f;
    float selfv = 0.0f;
    const float4*       pwp = pw + bn * 64;
    const unsigned int* pbp = pbin + bn * 64;
    for (int s = 0; s < 64; ++s) {
        float fv = fmaxf(src[(size_t)(b * 64 + s) * BD + tid], 0.0f);
        if (s == n) selfv = fv;
        float4 w = pwp[s];
        unsigned int pb = pbp[s];
        acc[(pb & 63u) * BD + tid]         += w.x * fv;
        acc[((pb >> 8) & 63u) * BD + tid]  += w.y * fv;
        acc[((pb >> 16) & 63u) * BD + tid] += w.z * fv;
        acc[((pb >> 24) & 63u) * BD + tid] += w.w * fv;
    }
    store_afrag<I>(Af, bn, tid, acc, selfv);
}

// ---------------------------------------------------------------------------
// WMMA GEMM: out[512 x ncols] = A[512 x 4*KST] * B[4*KST x 16*gridDim.y]
// One wave per 16x16 output tile; 4 independent fp32 accumulator chains
// (k strided by 4 fragment-steps) summed at the end for ILP.
// mode: 0 = store, 1 = store + residual, 2 = relu-store.
// ---------------------------------------------------------------------------
template <int KST>
__global__ void k_gemm(const float* __restrict__ Af, const float* __restrict__ Bf,
                       const float* __restrict__ resid, float* __restrict__ out,
                       int mode, int ncols) {
    int mtile = blockIdx.x, ntile = blockIdx.y;
    int lane = threadIdx.x;
    const v2f* A2 = (const v2f*)Af + (size_t)mtile * KST * 32 + lane;
    const v2f* B2 = (const v2f*)Bf + (size_t)ntile * KST * 32 + lane;
    v8f c0 = {0.f, 0.f, 0.f, 0.f, 0.f, 0.f, 0.f, 0.f};
    v8f c1 = c0, c2 = c0, c3 = c0;
    #pragma unroll 1
    for (int kk = 0; kk < KST; kk += 4) {
        v2f a0 = A2[0],  b0 = B2[0];
        v2f a1 = A2[32], b1 = B2[32];
        v2f a2 = A2[64], b2 = B2[64];
        v2f a3 = A2[96], b3 = B2[96];
        c0 = __builtin_amdgcn_wmma_f32_16x16x4_f32(false, a0, false, b0, (short)0, c0, false, false);
        c1 = __builtin_amdgcn_wmma_f32_16x16x4_f32(false, a1, false, b1, (short)0, c1, false, false);
        c2 = __builtin_amdgcn_wmma_f32_16x16x4_f32(false, a2, false, b2, (short)0, c2, false, false);
        c3 = __builtin_amdgcn_wmma_f32_16x16x4_f32(false, a3, false, b3, (short)0, c3, false, false);
        A2 += 128; B2 += 128;
    }
    v8f c = (c0 + c1) + (c2 + c3);
    int col = ntile * 16 + (lane & 15);
    int rbase = mtile * 16 + ((lane < 16) ? 0 : 8);
    #pragma unroll
    for (int j = 0; j < 8; ++j) {
        int row = rbase + j;
        float v = c[j];
        if (mode == 1) v += resid[row * 64 + col];
        if (mode == 2) v = fmaxf(v, 0.0f);
        if (col < ncols) out[(size_t)row * ncols + col] = v;
    }
}

// ---------------------------------------------------------------------------
// Fused per-step feature update (steps >= 1), one WG (256 threads) per (b,n):
//  (a) threads 0..127: feat[:, :16, :] *= tanh(equilinear(Wd_back, dec))
//  (b) all threads:    accumulate 48x2 weighted map features + count (LDS)
//  (c) threads 0..63:  contract with K_map, relu -> feat[:, 16:24, :]
// ---------------------------------------------------------------------------
__global__ void k_featmap(const float* __restrict__ dec, const float* __restrict__ Wdb,
                          const float* __restrict__ mp, const float* __restrict__ mf,
                          const float* __restrict__ mmask, const float* __restrict__ p,
                          const float* __restrict__ Kmap, float* __restrict__ featb) {
    __shared__ float W[NBIN * 2];
    __shared__ float cntS;
    int bn = blockIdx.x;
    int b = bn >> 6;
    int tid = threadIdx.x;
    if (tid < NBIN * 2) W[tid] = 0.0f;
    if (tid == 0) cntS = 0.0f;

    if (tid < 128) {                                  // (a) backprojection
        int o = tid >> 3, m = tid & 7;
        float s = 0.0f;
        #pragma unroll
        for (int i = 0; i < 3; ++i)
            #pragma unroll
            for (int l = 0; l < 8; ++l)
                s += Wdb[(o * 3 + i) * 8 + l] * dec[(size_t)bn * 24 + i * 8 + ((m - l) & 7)];
        featb[(size_t)bn * 192 + o * 8 + m] *= tanhf(s);
    }
    __syncthreads();

    float px = p[bn * 2], py = p[bn * 2 + 1];         // (b) map aggregation
    float lc = 0.0f;
    for (int s = tid; s < 1024; s += 256) {
        float rx = mp[(b * 1024 + s) * 2] - px;
        float ry = mp[(b * 1024 + s) * 2 + 1] - py;
        float d = sqrtf(rx * rx + ry * ry + 1e-12f);
        float valid = (d < RADIUS_F ? 1.0f : 0.0f) * mmask[b * 1024 + s];
        if (valid != 0.0f) {
            lc += valid;
            Corners c = polar_corners(rx, ry, d);
            float bw = valid * c.win;
            float f0 = mf[(b * 1024 + s) * 2];
            float f1 = mf[(b * 1024 + s) * 2 + 1];
            #pragma unroll
            for (int k = 0; k < 4; ++k) {
                float w = bw * c.cw[k];
                atomicAdd(&W[c.bin[k] * 2 + 0], w * f0);
                atomicAdd(&W[c.bin[k] * 2 + 1], w * f1);
            }
        }
    }
    atomicAdd(&cntS, lc);
    __syncthreads();

    if (tid < 64) {                                   // (c) map encoder output
        int o = tid >> 3, m = tid & 7;
        float inv = 1.0f / fmaxf(cntS, 1.0f);
        float e = 0.0f;
        for (int bin = 0; bin < NBIN; ++bin) {
            const float* kp = Kmap + ((size_t)(bin * 8 + o) * 8 + m) * 2;
            e += (W[bin * 2] * kp[0] + W[bin * 2 + 1] * kp[1]) * inv;
        }
        featb[(size_t)bn * 192 + (16 + o) * 8 + m] = fmaxf(e, 0.0f);
    }
}

// ---------------------------------------------------------------------------
// rho1 projection + trajectory output + position update (in place)
// ---------------------------------------------------------------------------
__global__ void k_post(const float* __restrict__ dec, const float* __restrict__ w_rho1,
                       float* __restrict__ pbuf, float* __restrict__ traj, int step) {
    int row = blockIdx.x * blockDim.x + threadIdx.x;
    if (row >= 512) return;
    float w0 = w_rho1[0], w1 = w_rho1[1];
    float dlt[3][2];
    #pragma unroll
    for (int c = 0; c < 3; ++c) {
        float d0 = 0.0f, d1 = 0.0f;
        #pragma unroll
        for (int m = 0; m < 8; ++m) {
            float ang = TWO_PI_F * (float)m * 0.125f;
            float cs = cosf(ang), sn = sinf(ang);
            float f = dec[(size_t)row * 24 + c * 8 + m];
            d0 += f * (w0 * cs - w1 * sn);
            d1 += f * (w0 * sn + w1 * cs);
        }
        dlt[c][0] = d0; dlt[c][1] = d1;
    }
    float pnx = pbuf[row * 2] + dlt[0][0];
    float pny = pbuf[row * 2 + 1] + dlt[0][1];
    pbuf[row * 2] = pnx; pbuf[row * 2 + 1] = pny;
    size_t base = ((size_t)step * 512 + row) * 6;
    traj[base + 0] = pnx;       traj[base + 1] = pny;
    traj[base + 2] = dlt[1][0]; traj[base + 3] = dlt[1][1];
    traj[base + 4] = dlt[2][0]; traj[base + 5] = dlt[2][1];
}

// ---------------------------------------------------------------------------
extern "C" void kernel_launch(void* const* d_in, const int* in_sizes, int n_in,
                              void* d_out, int out_size, void* d_ws, size_t ws_size,
                              hipStream_t stream) {
    const float* p      = (const float*)d_in[0];
    const float* feat   = (const float*)d_in[1];
    const float* map_p  = (const float*)d_in[2];
    const float* map_f  = (const float*)d_in[3];
    const float* car_m  = (const float*)d_in[4];
    const float* map_m  = (const float*)d_in[5];
    const float* K_v    = (const float*)d_in[6];
    const float* Wd_v   = (const float*)d_in[7];
    const float* K_1    = (const float*)d_in[8];
    const float* Wd_1   = (const float*)d_in[9];
    const float* K_2    = (const float*)d_in[10];
    const float* Wd_2   = (const float*)d_in[11];
    const float* Wd_b   = (const float*)d_in[12];
    const float* w_rho1 = (const float*)d_in[13];
    const float* K_map  = (const float*)d_in[14];
    float* traj = (float*)d_out;

    // workspace carve-up (all offsets 16-float aligned), ~24.3 MB total
    float* ws   = (float*)d_ws;
    float* K2v  = ws;                   // 4*2352*64  = 602112
    float* K2a  = K2v + 602112;         // 4*784*64   = 200704
    float* K2b  = K2a + 200704;         // 2*784*64   = 100352
    float* Af   = K2b + 100352;         // 32*2352*64 = 4816896
    float* out0 = Af + 4816896;         // 512*64
    float* out1 = out0 + 32768;         // 512*64
    float* dec  = out1 + 32768;         // 512*24
    float* pbuf = dec + 12288;          // 512*2
    float* featb = pbuf + 1024;         // 512*24*8
    float4*       pw   = (float4*)(featb + 98304);        // 512*64 float4
    unsigned int* pbin = (unsigned int*)(featb + 98304 + 131072);

    k_init<<<384, 256, 0, stream>>>(p, feat, pbuf, featb);
    k_bfrag<<<(4 * 2352 * 32) / 256, 256, 0, stream>>>(K_v, Wd_v, K2v, 24, 8, 2352, 4 * 2352 * 32);
    k_bfrag<<<(4 * 784 * 32) / 256, 256, 0, stream>>>(K_1, Wd_1, K2a, 8, 8, 784, 4 * 784 * 32);
    k_bfrag<<<(2 * 784 * 32) / 256, 256, 0, stream>>>(K_2, Wd_2, K2b, 8, 3, 784, 2 * 784 * 32);

    dim3 g4(32, 4), g2(32, 2);
    for (int t = 0; t < 30; ++t) {
        if (t > 0)
            k_featmap<<<512, 256, 0, stream>>>(dec, Wd_b, map_p, map_f, map_m,
                                               pbuf, K_map, featb);
        // layer 0: conv(K_vehicle) + equilinear(Wd_vehicle)   (pairs fused)
        k_agg0<<<512, 192, 0, stream>>>(featb, pbuf, car_m, pw, pbin, Af);
        k_gemm<2352><<<g4, 32, 0, stream>>>(Af, K2v, nullptr, out0, 0, 64);
        // layer 1: conv(K_conv1)+equi(Wd1) on relu(out0), + residual out0
        k_agg<8><<<512, 64, 0, stream>>>(out0, pw, pbin, Af);
        k_gemm<784><<<g4, 32, 0, stream>>>(Af, K2a, out0, out1, 1, 64);
        // layer 2: conv(K_conv2)+equi(Wd2) on relu(out1), relu -> dec [512,24]
        k_agg<8><<<512, 64, 0, stream>>>(out1, pw, pbin, Af);
        k_gemm<784><<<g2, 32, 0, stream>>>(Af, K2b, nullptr, dec, 2, 24);
        k_post<<<2, 256, 0, stream>>>(dec, w_rho1, pbuf, traj, t);
    }
    (void)in_sizes; (void)n_in; (void)out_size; (void)ws_size;
}